// LegendreProjectionUnit_26817775796704
// MI455X (gfx1250) — compile-verified
//
#include <hip/hip_runtime.h>
#include <hip/hip_bf16.h>

// CDNA5 / gfx1250, wave32. V_WMMA_F32_16X16X4_F32 for the f32 state recurrence.
// Round-1 asm showed (a) a fully serialized 16-deep WMMA accumulator chain with
// s_delay_alu stalls and (b) s_wait_storecnt_dscnt 0 before every barrier (waiting
// on global output stores). This round: 4 independent WMMA chains + an LDS-only
// split barrier so output stores stay in flight.

typedef float v2f __attribute__((ext_vector_type(2)));
typedef float v4f __attribute__((ext_vector_type(4)));
typedef float v8f __attribute__((ext_vector_type(8)));

#define NSTATE  64
#define LSEQ    512
#define DDIM    128
#define CSTRIDE 68   // padded LDS row stride (floats) to spread banks

// Workgroup barrier that only waits for LDS traffic (DScnt), not global STOREcnt.
// The per-step output stores are fire-and-forget into L2; nothing reads them.
__device__ __forceinline__ void wg_barrier_lds_only() {
    asm volatile(
        "s_wait_dscnt 0x0\n\t"
        "s_barrier_signal -1\n\t"
        "s_barrier_wait -1"
        ::: "memory");
}

// Grid: 128 blocks (= 16 B * 8 d-groups), 128 threads (4 wave32).
// Block g handles (b = g/8, d in [16*(g%8), 16*(g%8)+16)): 16 "p" rows.
// Wave w computes state columns m in [16w, 16w+16).
__launch_bounds__(128, 1)
__global__ void legt_scan_wmma_kernel(const float* __restrict__ inputs, // [16,512,128]
                                      const float* __restrict__ A,      // [64,64]
                                      const float* __restrict__ Bvec,   // [64]
                                      float* __restrict__ out)          // [16,128,64,512]
{
    __shared__ float fbuf[LSEQ * 16];          // staged inputs for this group (32 KB)
    __shared__ float cbuf[2][16 * CSTRIDE];    // double-buffered state c[p][n] (~8.5 KB)

    const int tid  = threadIdx.x;              // 0..127
    const int lane = tid & 31;                 // wave32 lane
    const int wave = tid >> 5;                 // 0..3 -> m-tile
    const int half = lane >> 4;                // 0/1 lane-half (WMMA K/M split)
    const int lm   = lane & 15;                // 0..15
    const int m    = wave * 16 + lm;           // state column 0..63

    const int gid = blockIdx.x;                // 0..127
    const int b   = gid >> 3;
    const int d0  = (gid & 7) * 16;

    // ---- stage f: fbuf[t*16 + p] = inputs[b, t, d0+p] (coalesced 64B bursts)
    const float* inBase = inputs + (size_t)b * LSEQ * DDIM + d0;
    for (int i = tid; i < LSEQ * 16; i += 128) {
        const int t = i >> 4, p = i & 15;
        fbuf[i] = inBase[(size_t)t * DDIM + p];
    }
    // ---- zero initial state (buffer 0 is read at t=0)
    for (int i = tid; i < 16 * CSTRIDE; i += 128) cbuf[0][i] = 0.0f;

    // ---- stationary B-operand: 16 chunks of A^T (B[n][m] = A[m][n]).
    // 4x16 f32 B-matrix layout: lane = column m; VGPR0 = row 4k+2h, VGPR1 = row 4k+2h+1.
    v2f bop[16];
#pragma unroll
    for (int k = 0; k < 16; ++k) {
        const float* ap = A + m * NSTATE + 4 * k + 2 * half;
        bop[k] = v2f{ap[0], ap[1]};
    }
    const float bv = Bvec[m];

    __syncthreads();   // full barrier once after init is fine

    // out[((gid*16 + p)*64 + m)*512 + t]
    float* outBase = out + (size_t)gid * 16 * NSTATE * LSEQ;

    for (int t = 0; t < LSEQ; ++t) {
        const int cur = t & 1, nxt = cur ^ 1;

        // acc0 = f[p] * Bvec[m]  (C/D layout: VGPR v, lane-half h -> p = v + 8h)
        const v4f f0 = *(const v4f*)&fbuf[t * 16 + 8 * half];
        const v4f f1 = *(const v4f*)&fbuf[t * 16 + 8 * half + 4];
        v8f acc0, acc1, acc2, acc3;
        acc0[0] = f0.x * bv; acc0[1] = f0.y * bv; acc0[2] = f0.z * bv; acc0[3] = f0.w * bv;
        acc0[4] = f1.x * bv; acc0[5] = f1.y * bv; acc0[6] = f1.z * bv; acc0[7] = f1.w * bv;
        acc1 = (v8f)(0.0f);
        acc2 = (v8f)(0.0f);
        acc3 = (v8f)(0.0f);

        // acc[p][m] += sum_n c[p][n] * A[m][n], K=64 as 4 independent chains of
        // 4 chained f32 WMMAs (breaks the D->C latency chain 4x).
        // A-operand (16x4 f32): lane = row p (= lm); VGPR0/1 = K 4k+2h / 4k+2h+1.
        const float* crow = &cbuf[cur][lm * CSTRIDE];
        v2f a[16];
#pragma unroll
        for (int k = 0; k < 16; ++k)
            a[k] = *(const v2f*)&crow[4 * k + 2 * half];

#pragma unroll
        for (int j = 0; j < 4; ++j) {
            acc0 = __builtin_amdgcn_wmma_f32_16x16x4_f32(false, a[j],      false, bop[j],      (short)0, acc0, false, false);
            acc1 = __builtin_amdgcn_wmma_f32_16x16x4_f32(false, a[4 + j],  false, bop[4 + j],  (short)0, acc1, false, false);
            acc2 = __builtin_amdgcn_wmma_f32_16x16x4_f32(false, a[8 + j],  false, bop[8 + j],  (short)0, acc2, false, false);
            acc3 = __builtin_amdgcn_wmma_f32_16x16x4_f32(false, a[12 + j], false, bop[12 + j], (short)0, acc3, false, false);
        }

        // reduce the 4 partial accumulators (elementwise, lowers to v_pk_add_f32)
        v8f acc;
#pragma unroll
        for (int v = 0; v < 8; ++v)
            acc[v] = (acc0[v] + acc1[v]) + (acc2[v] + acc3[v]);

        // publish new state + stream outputs (L2 write-combines the +t stream)
#pragma unroll
        for (int v = 0; v < 8; ++v) {
            const int p = v + 8 * half;
            cbuf[nxt][p * CSTRIDE + m] = acc[v];
            outBase[(size_t)(p * NSTATE + m) * LSEQ + t] = acc[v];
        }
        // LDS-only split barrier: don't wait for the global output stores.
        wg_barrier_lds_only();
    }
}

extern "C" void kernel_launch(void* const* d_in, const int* in_sizes, int n_in,
                              void* d_out, int out_size, void* d_ws, size_t ws_size,
                              hipStream_t stream) {
    const float* inputs = (const float*)d_in[0];  // [16,512,128]
    const float* A      = (const float*)d_in[1];  // [64,64]
    const float* Bvec   = (const float*)d_in[2];  // [64]
    float* out          = (float*)d_out;          // [16,128,64,512]
    (void)in_sizes; (void)n_in; (void)out_size; (void)d_ws; (void)ws_size;

    legt_scan_wmma_kernel<<<dim3(128), dim3(128), 0, stream>>>(inputs, A, Bvec, out);
}